// GSU_64982855188968
// MI455X (gfx1250) — compile-verified
//
#include <hip/hip_runtime.h>
#include <stdint.h>

// GSU soft-retrieval for MI455X (gfx1250, wave32).
// B=1024, L=4096, D=64, K=64. Outputs: retrieved_emb [B,K,D] f32 then
// retrieved_mask [B,K] (as 0/1 f32), concatenated flat in d_out.
//
// Design: HBM-bound (1 GiB of seq_emb per call, ~46us floor @ 23.3 TB/s).
// Scores computed on the f32 WMMA path (v_wmma_f32_16x16x4_f32); top-64 via
// iterative argmax on packed u64 keys in LDS; coalesced gather at the end.

typedef __attribute__((ext_vector_type(2))) float v2f;
typedef __attribute__((ext_vector_type(8))) float v8f;

#define B_ 1024
#define L_ 4096
#define D_ 64
#define K_ 64
#define NEGV (-1e9f)

// ---------------------------------------------------------------------------
// Kernel 1: q = target_emb @ W^T   ([B,64] = [B,64] x [64,64]^T)
// One wave per 16x16 output tile; 16 chained v_wmma_f32_16x16x4_f32 over K=64.
// WMMA f32 16x16x4 operand layouts (ISA 7.12.2):
//   A (16x4): lanes 0-15 hold row M=lane, VGPR0=K0,VGPR1=K1; lanes16-31 K2,K3.
//   B (4x16): VGPR v: lanes0-15 -> row K=v, lanes16-31 -> row K=v+2 (N=lane&15).
//   C (16x16): VGPR v: lanes0-15 M=v, lanes16-31 M=v+8, N=lane&15.
// ---------------------------------------------------------------------------
__global__ __launch_bounds__(256) void gsu_proj_q(const float* __restrict__ target,
                                                  const float* __restrict__ W,
                                                  float* __restrict__ qout) {
  const int lane  = threadIdx.x & 31;
  const int wave  = blockIdx.x * 8 + (threadIdx.x >> 5);
  const int tilesN = D_ / 16;                 // 4
  const int m0 = (wave / tilesN) * 16;        // batch-row base
  const int n0 = (wave % tilesN) * 16;        // output-column base
  const int lane15 = lane & 15;               // M for A, N for B/C
  const int grp    = lane >> 4;               // selects K pair {0,1} vs {2,3}

  v8f c = {};
  #pragma unroll
  for (int kb = 0; kb < D_; kb += 4) {
    // A fragment: target[m0+lane15][kb + 2*grp .. +1]
    v2f a = *(const v2f*)(target + (m0 + lane15) * D_ + kb + 2 * grp);
    // B fragment: B[k][n] = (W^T)[k][n] = W[n0+lane15][kb + 2*grp .. +1]
    v2f bb = *(const v2f*)(W + (n0 + lane15) * D_ + kb + 2 * grp);
    c = __builtin_amdgcn_wmma_f32_16x16x4_f32(false, a, false, bb,
                                              (short)0, c, false, false);
  }
  #pragma unroll
  for (int v = 0; v < 8; ++v) {
    const int row = m0 + v + 8 * grp;
    qout[row * D_ + n0 + lane15] = c[v];
  }
}

// ---------------------------------------------------------------------------
// Kernel 2: per-batch scores (WMMA) + top-64 select + gather.
// One block (8 waves / 256 threads) per batch row.
// ---------------------------------------------------------------------------
__global__ __launch_bounds__(256) void gsu_scores_topk(const float* __restrict__ seq,
                                                       const int*   __restrict__ mask,
                                                       const float* __restrict__ q,
                                                       float* __restrict__ out_emb,
                                                       float* __restrict__ out_mask) {
  __shared__ float s_q[D_];
  __shared__ float s_sc[L_];                         // 16 KB of masked scores
  __shared__ unsigned long long s_key[256];
  __shared__ int   s_idx[K_];
  __shared__ float s_val[K_];

  const int b    = blockIdx.x;
  const int tid  = threadIdx.x;
  const int lane = tid & 31;
  const int wv   = tid >> 5;                         // 0..7
  const float* seqb  = seq  + (size_t)b * L_ * D_;
  const int*   maskb = mask + (size_t)b * L_;

  if (tid < D_) s_q[tid] = q[b * D_ + tid];
  __syncthreads();

  const int lane15 = lane & 15;
  const int grp    = lane >> 4;

  // ---- scores: each wave owns 16 L-rows per chunk of 128 ----
  for (int l0 = wv * 16; l0 < L_; l0 += 128) {
    const float* rowp = seqb + (size_t)(l0 + lane15) * D_ + 2 * grp;
    v8f c = {};
    #pragma unroll
    for (int kb = 0; kb < D_; kb += 4) {
      v2f a = *(const v2f*)(rowp + kb);              // A: seq rows
      v2f bb;                                        // B: q replicated over N
      bb.x = s_q[kb + 2 * grp];
      bb.y = s_q[kb + 2 * grp + 1];
      c = __builtin_amdgcn_wmma_f32_16x16x4_f32(false, a, false, bb,
                                                (short)0, c, false, false);
    }
    if (l0 + 128 < L_)                               // stream-ahead hint
      __builtin_prefetch(seqb + (size_t)(l0 + 128 + lane15) * D_ + 2 * grp, 0, 1);
    // Every column of the tile equals the row score; lanes with N==0 extract.
    if (lane15 == 0) {
      #pragma unroll
      for (int v = 0; v < 8; ++v) {
        const int row = l0 + v + 8 * grp;
        const float sc = c[v] * 0.125f;              // 1/sqrt(64)
        s_sc[row] = maskb[row] ? sc : NEGV;
      }
    }
  }
  __syncthreads();

  // ---- top-64: 64 rounds of block-wide argmax on packed u64 keys ----
  for (int r = 0; r < K_; ++r) {
    unsigned long long best = 0ull;
    for (int i = tid; i < L_; i += 256) {
      unsigned u = __float_as_uint(s_sc[i]);
      u ^= (unsigned)((int)u >> 31) | 0x80000000u;   // order-preserving map
      const unsigned long long key =
          ((unsigned long long)u << 32) | (unsigned)(L_ - 1 - i);  // tie->low idx
      best = (key > best) ? key : best;
    }
    s_key[tid] = best;
    __syncthreads();
    if (wv == 0) {
      unsigned long long m = s_key[lane];
      #pragma unroll
      for (int j = 1; j < 8; ++j) {
        const unsigned long long t = s_key[lane + 32 * j];
        m = (t > m) ? t : m;
      }
      #pragma unroll
      for (int off = 16; off > 0; off >>= 1) {
        const unsigned long long t = __shfl_xor(m, off, 32);
        m = (t > m) ? t : m;
      }
      if (lane == 0) {
        const int idx = L_ - 1 - (int)(m & 0xFFFFFFFFu);
        s_idx[r] = idx;
        s_val[r] = s_sc[idx];
        s_sc[idx] = -3.0e38f;                        // knock out (< NEGV)
      }
    }
    __syncthreads();
  }

  // ---- gather retrieved_emb [K,D] (coalesced over D) + mask ----
  for (int i = tid; i < K_ * D_; i += 256) {
    const int k = i >> 6, d = i & 63;
    out_emb[(size_t)b * K_ * D_ + i] = seqb[(size_t)s_idx[k] * D_ + d];
  }
  if (tid < K_)
    out_mask[(size_t)b * K_ + tid] = (s_val[tid] > -1e8f) ? 1.0f : 0.0f;
}

// ---------------------------------------------------------------------------
// Inputs (setup_inputs order):
//   0 target_emb [B,D] f32, 1 target_category [B] i32, 2 seq_emb [B,L,D] f32,
//   3 seq_category [B,L] i32, 4 seq_mask [B,L] int, 5 top_k scalar, 6 W [D,D] f32
// ---------------------------------------------------------------------------
extern "C" void kernel_launch(void* const* d_in, const int* in_sizes, int n_in,
                              void* d_out, int out_size, void* d_ws, size_t ws_size,
                              hipStream_t stream) {
  const float* target = (const float*)d_in[0];
  const float* seq    = (const float*)d_in[2];
  const int*   mask   = (const int*)d_in[4];
  const float* W      = (const float*)d_in[6];

  float* q        = (float*)d_ws;                        // B*D floats = 256 KiB
  float* out_emb  = (float*)d_out;                       // [B,K,D]
  float* out_mask = out_emb + (size_t)B_ * K_ * D_;      // [B,K]

  // 256 tiles total, 8 waves/block -> 32 blocks
  gsu_proj_q<<<(B_ / 16) * (D_ / 16) / 8, 256, 0, stream>>>(target, W, q);
  gsu_scores_topk<<<B_, 256, 0, stream>>>(seq, mask, q, out_emb, out_mask);
}